// ClassCapsule_58540404244861
// MI455X (gfx1250) — compile-verified
//
#include <hip/hip_runtime.h>

// ---------------------------------------------------------------------------
// Matrix-capsule EM routing (ClassCapsule) for MI455X / gfx1250.
//  Kernel 1: votes = pose(4x4) @ w(4x4) via V_WMMA_F32_16X16X4_F32.
//            One wave: A = (4 spatial positions x 4 pose rows) for one input
//            capsule i, then 3 unrolled WMMAs over the o-tiles (O=10 -> 12),
//            A reused from registers.
//  Kernel 2: fused 3-iteration EM routing, one 1024-thread workgroup per
//            batch element; votes (188 MB) stay resident in the 192 MB L2.
// ---------------------------------------------------------------------------

typedef float v2f __attribute__((ext_vector_type(2)));
typedef float v8f __attribute__((ext_vector_type(8)));

#define EPS        1e-9f
#define Bsz        64
#define Hh         12
#define Ww         12
#define Ii         32
#define Oo         10
#define Pp         16
#define HW         (Hh * Ww)        /* 144  */
#define Nn         (HW * Ii)        /* 4608 */
#define NG         (HW / 4)         /* 36 groups of 4 positions */
#define OT         3                /* o-tiles of 4 (O=10 padded to 12) */

// ---------------------------------------------------------------------------
// Kernel 1: votes[b][n][o][p]  (n = hw*I + i), with coordinate addition.
// ---------------------------------------------------------------------------
__global__ __launch_bounds__(256)
void votes_wmma_kernel(const float* __restrict__ pose,   // [B][H][W][I][4][4]
                       const float* __restrict__ w,      // [I][O][4][4]
                       float* __restrict__ votes)        // [B][N][O][16]
{
    const int gwave = (blockIdx.x * blockDim.x + threadIdx.x) >> 5;
    const int lane  = threadIdx.x & 31;

    int task = gwave;                       // (b*NG + g)*I + i
    const int i  = task % Ii; task /= Ii;
    const int g  = task % NG; task /= NG;
    const int b  = task;
    if (b >= Bsz) return;

    const int half = lane >> 4;             // 0 -> K={0,1}, 1 -> K={2,3}
    const int m    = lane & 15;

    // ---- A (16x4 f32): row M = nn*4 + a  (nn = position in group, a = pose row)
    const int nn  = m >> 2, a = m & 3;
    const int hwp = g * 4 + nn;
    const float* pa = pose + ((((size_t)b * HW + hwp) * Ii + i) * 4 + a) * 4 + 2 * half;
    const v2f A = *(const v2f*)pa;          // 8B-aligned -> global_load_b64

    // ---- lane's D-column decode: col N = oo*4 + c
    const int oo = m >> 2, c = m & 3;

    // ---- hoisted per-row store bases + coordinate offsets (ot-invariant).
    // D VGPR v holds row M = v + 8*half, column = m; p = a2*4 + c.
    constexpr float invHW = 1.0f / 12.0f;
    float  offv[8];
    size_t basev[8];
    #pragma unroll
    for (int v = 0; v < 8; ++v) {
        const int row = v + 8 * half;
        const int nn2 = row >> 2, a2 = row & 3;
        const int hw2 = g * 4 + nn2;
        const int hid = hw2 / Ww;
        const int wid = hw2 - hid * Ww;
        const int p   = a2 * 4 + c;
        offv[v]  = (p == 0) ? ((float)hid + 0.5f) * invHW
                 : (p == 1) ? ((float)wid + 0.5f) * invHW : 0.0f;
        basev[v] = ((size_t)b * Nn + (size_t)hw2 * Ii + i) * (Oo * Pp) + p;
    }

    // ---- 3 WMMAs per wave, A reused from registers
    #pragma unroll
    for (int ot = 0; ot < OT; ++ot) {
        const int o = ot * 4 + oo;
        v2f Bv; Bv[0] = 0.0f; Bv[1] = 0.0f;
        if (o < Oo) {
            const float* pb = w + (((size_t)i * Oo + o) * 4 + 2 * half) * 4 + c;
            Bv[0] = pb[0];          // k = 2*half
            Bv[1] = pb[4];          // k = 2*half + 1
        }
        v8f C = {};
        C = __builtin_amdgcn_wmma_f32_16x16x4_f32(
                /*neg_a=*/false, A, /*neg_b=*/false, Bv,
                /*c_mod=*/(short)0, C, /*reuse_a=*/false, /*reuse_b=*/false);

        if (o < Oo) {
            #pragma unroll
            for (int v = 0; v < 8; ++v)
                votes[basev[v] + (size_t)o * Pp] = C[v] + offv[v];
        }
    }
}

// ---------------------------------------------------------------------------
// Kernel 2: fused EM routing, one block (1024 thr = 32 waves) per batch elem.
// Half-wave owns one n; lane = component p. rr recomputed on the fly from
// previous iteration's per-o stats held in LDS (E-step fused into M-step).
// ---------------------------------------------------------------------------
__global__ __launch_bounds__(1024)
void em_routing_kernel(const float* __restrict__ votes,  // [B][N][O][16]
                       const float* __restrict__ iact,   // [B][N]
                       const float* __restrict__ beta_v, // [O]
                       const float* __restrict__ beta_a, // [O]
                       float* __restrict__ out)          // pose(B*O*16) ++ act(B*O)
{
    const int b = blockIdx.x;

    __shared__ float s_mean[Oo][Pp], s_i2v[Oo][Pp], s_lstd[Oo][Pp];
    __shared__ float s_logact[Oo], s_act[Oo];
    __shared__ float s_rv[Oo][Pp], s_rv2[Oo][Pp], s_rs[Oo];
    __shared__ float s_cost[Oo];

    const int tid   = threadIdx.x;
    const int wave  = tid >> 5;
    const int lane  = tid & 31;
    const int half  = lane >> 4;
    const int p     = lane & 15;
    const int nstep = (blockDim.x >> 5) * 2;             // n's per pass = 64

    const float* V  = votes + (size_t)b * Nn * Oo * Pp;
    const float* IA = iact  + (size_t)b * Nn;

    // inv_temp = 0.01 * (1 - 0.95^(it+1))
    const float INVT[3] = {0.01f * 0.05f, 0.01f * 0.0975f, 0.01f * 0.142625f};

    for (int it = 0; it < 3; ++it) {
        // ---- zero accumulators
        for (int t = tid; t < Oo * Pp; t += blockDim.x) {
            (&s_rv[0][0])[t]  = 0.0f;
            (&s_rv2[0][0])[t] = 0.0f;
        }
        if (tid < Oo) s_rs[tid] = 0.0f;
        __syncthreads();

        float a_rv[Oo], a_rv2[Oo], a_rs[Oo];
        #pragma unroll
        for (int o = 0; o < Oo; ++o) { a_rv[o] = 0.f; a_rv2[o] = 0.f; a_rs[o] = 0.f; }

        #pragma unroll 2
        for (int n = wave * 2 + half; n < Nn; n += nstep) {
            float v[Oo];
            #pragma unroll
            for (int o = 0; o < Oo; ++o)
                v[o] = V[((size_t)n * Oo + o) * Pp + p];

            float rr[Oo];
            if (it == 0) {
                #pragma unroll
                for (int o = 0; o < Oo; ++o) rr[o] = 1.0f / (float)Oo;
            } else {
                float zz[Oo];
                #pragma unroll
                for (int o = 0; o < Oo; ++o) {
                    const float d = v[o] - s_mean[o][p];
                    float t = d * d * s_i2v[o][p] + s_lstd[o][p];
                    // sum over the 16 components (butterfly within half-wave)
                    #pragma unroll
                    for (int off = 8; off >= 1; off >>= 1)
                        t += __shfl_xor(t, off, 16);
                    zz[o] = s_logact[o] - t;   // log(act+eps) + o_p0 + o_p2
                }
                // softmax over o (all lanes of the half hold identical zz)
                float mx = zz[0];
                #pragma unroll
                for (int o = 1; o < Oo; ++o) mx = fmaxf(mx, zz[o]);
                float s = 0.0f;
                #pragma unroll
                for (int o = 0; o < Oo; ++o) { rr[o] = __expf(zz[o] - mx); s += rr[o]; }
                const float inv = __builtin_amdgcn_rcpf(s);   // v_rcp_f32 in hot loop
                #pragma unroll
                for (int o = 0; o < Oo; ++o) rr[o] *= inv;
            }

            const float ia = IA[n];
            #pragma unroll
            for (int o = 0; o < Oo; ++o) {
                const float rp = rr[o] * ia;
                a_rs[o]  += rp;
                a_rv[o]  += rp * v[o];
                a_rv2[o] += rp * v[o] * v[o];
            }
        }

        // ---- merge per-lane partials (lane == component p) via ds_add_f32
        #pragma unroll
        for (int o = 0; o < Oo; ++o) {
            atomicAdd(&s_rv[o][p],  a_rv[o]);
            atomicAdd(&s_rv2[o][p], a_rv2[o]);
            if (p == 0) atomicAdd(&s_rs[o], a_rs[o]);
        }
        __syncthreads();

        // ---- per-(o,p) statistics (one-off work, 160 threads)
        if (tid < Oo * Pp) {
            const int o = tid >> 4, pp = tid & 15;
            const float rs   = s_rs[o];
            const float irs  = __builtin_amdgcn_rcpf(rs);
            const float mean = s_rv[o][pp] * irs;
            const float var  = fmaxf(s_rv2[o][pp] * irs - mean * mean, 0.0f);
            const float stdv = __builtin_sqrtf(var);
            s_mean[o][pp] = mean;
            s_i2v[o][pp]  = 0.5f * __builtin_amdgcn_rcpf(var); // 1/(2 sigma^2)
            s_lstd[o][pp] = __logf(stdv + EPS);
        }
        __syncthreads();

        // ---- per-o cost
        if (tid < Oo) {
            const float bv = beta_v[tid];
            const float rs = s_rs[tid];
            float c = 0.0f;
            #pragma unroll
            for (int pp = 0; pp < Pp; ++pp) c += (bv + s_lstd[tid][pp]) * rs;
            s_cost[tid] = c;
        }
        __syncthreads();

        // ---- activations (tiny serial work on thread 0)
        if (tid == 0) {
            float cm = 0.0f;
            for (int o = 0; o < Oo; ++o) cm += s_cost[o];
            cm /= (float)Oo;
            float cv = 0.0f;
            for (int o = 0; o < Oo; ++o) { const float d = s_cost[o] - cm; cv += d * d; }
            const float cs = __builtin_sqrtf(cv / (float)Oo);
            const float invt = INVT[it];
            for (int o = 0; o < Oo; ++o) {
                const float ac  = beta_a[o] + (cm - s_cost[o]) / (cs + EPS);
                const float act = 1.0f / (1.0f + __expf(-invt * ac));
                s_act[o]    = act;
                s_logact[o] = __logf(act + EPS);
            }
        }
        __syncthreads();
    }

    // ---- outputs: pose (B,O,4,4) then activation (B,O), both f32
    if (tid < Oo * Pp) out[(size_t)b * Oo * Pp + tid] = s_mean[tid >> 4][tid & 15];
    if (tid < Oo)      out[(size_t)Bsz * Oo * Pp + (size_t)b * Oo + tid] = s_act[tid];
}

// ---------------------------------------------------------------------------
extern "C" void kernel_launch(void* const* d_in, const int* in_sizes, int n_in,
                              void* d_out, int out_size, void* d_ws, size_t ws_size,
                              hipStream_t stream) {
    const float* pose   = (const float*)d_in[0];  // (64,12,12,32,4,4)
    const float* iact   = (const float*)d_in[1];  // (64,12,12,32)
    const float* w      = (const float*)d_in[2];  // (1,32,10,4,4)
    const float* beta_v = (const float*)d_in[3];  // (1,10)
    const float* beta_a = (const float*)d_in[4];  // (1,10)
    float*       out    = (float*)d_out;          // 10240 pose + 640 act
    float*       votes  = (float*)d_ws;           // 64*4608*10*16 f32 = 188 MB

    // Kernel 1: 73728 wave-tasks (3 WMMAs each), 8 waves per 256-thread block
    const int tasks  = Bsz * NG * Ii;
    const int blocks = tasks / 8;                 // 9216
    votes_wmma_kernel<<<blocks, 256, 0, stream>>>(pose, w, votes);

    // Kernel 2: one max-size workgroup (32 waves) per batch element
    em_routing_kernel<<<Bsz, 1024, 0, stream>>>(votes, iact, beta_v, beta_a, out);
}